// BestAnchor_34007551050568
// MI455X (gfx1250) — compile-verified
//
#include <hip/hip_runtime.h>
#include <stdint.h>

// Problem constants from the reference
#define BB 16
#define NN 100000
#define MM 32

#define BPB 64        // blocks per batch
#define TILE 256      // anchors staged per tile
#define THREADS 256   // 8 waves of 32

typedef unsigned long long u64;
typedef int v4i __attribute__((ext_vector_type(4)));

#define AS1 __attribute__((address_space(1)))
#define AS3 __attribute__((address_space(3)))

#if defined(__has_builtin)
# if __has_builtin(__builtin_amdgcn_global_load_async_to_lds_b128) && \
     __has_builtin(__builtin_amdgcn_global_load_async_to_lds_b32)
#  define HAVE_ASYNC_LDS 1
# endif
# if __has_builtin(__builtin_amdgcn_s_wait_asynccnt)
#  define HAVE_WAIT_ASYNC 1
# endif
#endif

__device__ __forceinline__ void async_b128(void* lds, const void* g) {
#ifdef HAVE_ASYNC_LDS
  __builtin_amdgcn_global_load_async_to_lds_b128(
      (AS1 v4i*)g, (AS3 v4i*)lds, 0, 0);
#else
  unsigned l = (unsigned)(uintptr_t)(AS3 void*)lds;
  asm volatile("global_load_async_to_lds_b128 %0, %1, off"
               :: "v"(l), "v"(g) : "memory");
#endif
}

__device__ __forceinline__ void async_b32(void* lds, const void* g) {
#ifdef HAVE_ASYNC_LDS
  __builtin_amdgcn_global_load_async_to_lds_b32(
      (AS1 int*)g, (AS3 int*)lds, 0, 0);
#else
  unsigned l = (unsigned)(uintptr_t)(AS3 void*)lds;
  asm volatile("global_load_async_to_lds_b32 %0, %1, off"
               :: "v"(l), "v"(g) : "memory");
#endif
}

__device__ __forceinline__ void wait_async_le2() {
#ifdef HAVE_WAIT_ASYNC
  __builtin_amdgcn_s_wait_asynccnt(2);
#else
  asm volatile("s_wait_asynccnt 0x2" ::: "memory");
#endif
}

__device__ __forceinline__ void wait_async_le0() {
#ifdef HAVE_WAIT_ASYNC
  __builtin_amdgcn_s_wait_asynccnt(0);
#else
  asm volatile("s_wait_asynccnt 0x0" ::: "memory");
#endif
}

// ---------------------------------------------------------------------------
// Main scan (placed first so the disasm snippet shows it).
// Lane m of each wave owns target m (wave32 == M). Anchors are wave-uniform:
// staged tile-by-tile via async global->LDS copies (double buffered on
// ASYNCcnt), then read back as conflict-free LDS broadcasts. Per-lane running
// best merged globally via atomic_max_u64 on packed (f32_bits<<32)|~index
// -> exact argmax with first-occurrence tie-break.
// ---------------------------------------------------------------------------
__global__ __launch_bounds__(THREADS) void bestanchor_main(
    const float*  __restrict__ score,
    const float4* __restrict__ bbox,
    const float4* __restrict__ target,
    u64*          __restrict__ ws) {
  __shared__ float4 sb[2][TILE];
  __shared__ float  ss[2][TILE];

  const int b    = blockIdx.x / BPB;
  const int blk  = blockIdx.x % BPB;
  const int tid  = threadIdx.x;
  const int lane = tid & 31;
  const int wave = tid >> 5;

  const int chunk  = (NN + BPB - 1) / BPB;         // 1563
  const int nStart = blk * chunk;
  const int nEnd   = (nStart + chunk < NN) ? (nStart + chunk) : NN;
  const int len    = nEnd - nStart;
  const int tiles  = (len + TILE - 1) / TILE;

  const float4* bb_base = bbox  + (size_t)b * NN;
  const float*  sc_base = score + (size_t)b * NN;

  auto stage = [&](int t) {
    int idx = nStart + t * TILE + tid;
    int lim = nEnd - 1;
    idx = (idx < lim) ? idx : lim;                 // clamp (dup loads harmless)
    int buf = t & 1;
    async_b128(&sb[buf][tid], (const void*)(bb_base + idx));
    async_b32 (&ss[buf][tid], (const void*)(sc_base + idx));
  };

  // Per-lane target (coalesced 16B load) + its area.
  float4 tg = target[b * MM + lane];
  const float areaT = (tg.z - tg.x) * (tg.w - tg.y);

  float    bestV = -1.0f;
  unsigned bestN = 0u;

  stage(0);
  if (tiles > 1) stage(1);

  for (int t = 0; t < tiles; ++t) {
    if (t + 1 < tiles) wait_async_le2();           // tile t landed (in-order)
    else               wait_async_le0();
    __syncthreads();                               // everyone's copies visible

    const int buf  = t & 1;
    const int base = nStart + t * TILE + wave * 32; // uniform per wave
    int cnt = nEnd - base;
    cnt = (cnt < 32) ? cnt : 32;

    // Prune threshold: wave-min of running bests. combined <= score, so any
    // anchor with score <= thresh cannot improve ANY lane's best.
    float tm = bestV;
    tm = fminf(tm, __shfl_xor(tm, 16));
    tm = fminf(tm, __shfl_xor(tm, 8));
    tm = fminf(tm, __shfl_xor(tm, 4));
    tm = fminf(tm, __shfl_xor(tm, 2));
    tm = fminf(tm, __shfl_xor(tm, 1));
    const float thresh = tm;                       // identical across lanes

    #pragma unroll 4
    for (int a = 0; a < cnt; ++a) {
      const int slot = wave * 32 + a;              // uniform -> LDS broadcast
      const float sc = ss[buf][slot];
      // Predicate is identical across lanes; ballot forces a true scalar
      // branch (s_cbranch) so pruned anchors skip the whole IoU body.
      if (__ballot(sc > thresh) != 0ull) {
        const float4 bv = sb[buf][slot];
        float ltx = fmaxf(bv.x, tg.x);
        float lty = fmaxf(bv.y, tg.y);
        float rbx = fminf(bv.z, tg.z);
        float rby = fminf(bv.w, tg.w);
        float w = fmaxf(rbx - ltx, 0.0f);
        float h = fmaxf(rby - lty, 0.0f);
        float inter = w * h;
        float areaB = (bv.z - bv.x) * (bv.w - bv.y);
        float uni = fmaxf(areaB + areaT - inter, 1e-6f);
        float v = sc * inter * __builtin_amdgcn_rcpf(uni);
        if (v > bestV) { bestV = v; bestN = (unsigned)(base + a); }
      }
    }

    __syncthreads();                               // buf free for re-staging
    if (t + 2 < tiles) stage(t + 2);
  }

  if (bestV >= 0.0f) {
    // combined >= 0 always => float bits order == uint order. ~idx makes
    // smaller index win on value ties (argmax first-occurrence semantics).
    u64 packed = ((u64)__float_as_uint(bestV) << 32) | (u64)(~bestN);
    atomicMax(&ws[b * MM + lane], packed);
  }
}

// ---------------------------------------------------------------------------
// Zero the 512 packed (value,index) reduction slots in workspace.
// ---------------------------------------------------------------------------
__global__ void bestanchor_init(u64* __restrict__ ws) {
  int i = threadIdx.x;
  if (i < BB * MM) ws[i] = 0ull;
}

// ---------------------------------------------------------------------------
// Unpack winners and gather their bboxes.
// ---------------------------------------------------------------------------
__global__ void bestanchor_final(const float4* __restrict__ bbox,
                                 const u64*    __restrict__ ws,
                                 float4*       __restrict__ out) {
  int i = blockIdx.x * blockDim.x + threadIdx.x;
  if (i < BB * MM) {
    int b = i >> 5;
    u64 p = ws[i];
    unsigned n = ~(unsigned)(p & 0xFFFFFFFFull);
    if (n >= NN) n = 0;                            // defensive
    out[i] = bbox[(size_t)b * NN + n];
  }
}

extern "C" void kernel_launch(void* const* d_in, const int* in_sizes, int n_in,
                              void* d_out, int out_size, void* d_ws, size_t ws_size,
                              hipStream_t stream) {
  (void)in_sizes; (void)n_in; (void)out_size; (void)ws_size;
  const float*  score  = (const float*)d_in[0];    // (B, N)
  const float4* bbox   = (const float4*)d_in[1];   // (B, N, 4)
  const float4* target = (const float4*)d_in[2];   // (B, M, 4)
  u64*    ws  = (u64*)d_ws;                        // 512 packed slots
  float4* out = (float4*)d_out;                    // (B, M, 4)

  hipLaunchKernelGGL(bestanchor_init, dim3(1), dim3(512), 0, stream, ws);
  hipLaunchKernelGGL(bestanchor_main, dim3(BB * BPB), dim3(THREADS), 0, stream,
                     score, bbox, target, ws);
  hipLaunchKernelGGL(bestanchor_final, dim3(2), dim3(256), 0, stream,
                     bbox, ws, out);
}